// Q_Decoder_3229815407376
// MI455X (gfx1250) — compile-verified
//
#include <hip/hip_runtime.h>
#include <cmath>

// ---------------------------------------------------------------------------
// Q_Decoder step for MI455X (gfx1250, wave32).
// Memory-bound GEMV pipeline (~26 MB streamed, ~1.1 us at 23.3 TB/s).
// GEMVs run as 16xNxK WMMA matmuls (V_WMMA_F32_16X16X4_F32) fed from LDS.
// Weight tiles are streamed HBM->LDS with GLOBAL_LOAD_ASYNC_TO_LDS_B128 and
// double-buffered per wave (ASYNCcnt in-order completion), falling back to
// synchronous b128 staging if the async builtins are not declared.
// w_hh is never read (h0 == 0 in the reference).
// ---------------------------------------------------------------------------

#define H      1024
#define MAXLEN 512

typedef float v2f __attribute__((ext_vector_type(2)));
typedef float v8f __attribute__((ext_vector_type(8)));

// exact parameter types of __builtin_amdgcn_global_load_async_to_lds_b128:
//   (v4i AS(1)*, v4i AS(3)*, imm int offset, imm int cpol)
typedef int v4i_gcc __attribute__((vector_size(16)));
typedef __attribute__((address_space(1))) v4i_gcc* as1_v4i_ptr;  // global
typedef __attribute__((address_space(3))) v4i_gcc* as3_v4i_ptr;  // LDS

#if defined(__AMDGCN__) &&                                                    \
    __has_builtin(__builtin_amdgcn_global_load_async_to_lds_b128) &&          \
    __has_builtin(__builtin_amdgcn_s_wait_asynccnt)
#define USE_ASYNC_LDS 1
#else
#define USE_ASYNC_LDS 0
#endif

// ---------------------------------------------------------------------------
// K1: c = [embed_row , hidden]  (2048)   and   x2[0:H] = embed_row
// ---------------------------------------------------------------------------
__global__ void build_vecs_kernel(const int* __restrict__ inputId,
                                  const float* __restrict__ hidden,
                                  const float* __restrict__ embed_table,
                                  float* __restrict__ c,
                                  float* __restrict__ x2) {
    int t = blockIdx.x * blockDim.x + threadIdx.x;   // 0..2047
    int id = inputId[0];
    if (t < H) {
        float v = embed_table[(size_t)id * H + t];
        c[t]  = v;
        x2[t] = v;
    } else {
        c[t] = hidden[t - H];
    }
}

// ---------------------------------------------------------------------------
// WMMA GEMV:  out[n] = act( sum_k W[n][k] * x[k] + bias[n] ),  W is N x K
// row-major. Block = 256 threads (8 waves); each wave produces 16 rows.
// Per 64-wide k-chunk: the 16x64 f32 tile goes HBM->LDS (async b128,
// double-buffered), then 16 x v_wmma_f32_16x16x4_f32 with
//   A[m][k] = W[r0+m][k0+k]        (lane m / lane 16+m hold the K pairs)
//   B[k][n] = x[k0+k] for all n    (columns identical -> 16 outputs per D)
// N must equal gridDim.x * 128. K must be a multiple of 64 (<= 2048).
// ---------------------------------------------------------------------------
template <int ACT_RELU>
__global__ void gemv_wmma_kernel(const float* __restrict__ W,
                                 const float* __restrict__ x,
                                 const float* __restrict__ bias,
                                 float* __restrict__ out,
                                 int K) {
    __shared__ __align__(16) float xs[2048];
#if USE_ASYNC_LDS
    __shared__ __align__(16) float wt[8][2][16 * 64];   // ping-pong per wave
#else
    __shared__ __align__(16) float wt[8][16 * 64];
#endif

    const int tid  = threadIdx.x;
    const int wave = tid >> 5;
    const int lane = tid & 31;

    // stage x into LDS (block-wide, coalesced)
    for (int i = tid; i < K; i += 256) xs[i] = x[i];
    __syncthreads();

    const int r0  = (blockIdx.x * 8 + wave) * 16;   // first output row of wave
    const int m   = lane & 15;                      // matrix row held by lane
    const int off = (lane >> 4) * 2;                // K-pair select (0 or 2)

    v8f acc = {};

#if USE_ASYNC_LDS
    float* Wnc = const_cast<float*>(W);
    // 8 async b128 loads move one 16x64 f32 tile (coalesced 256B/row)
#define ISSUE_TILE(K0, BUF)                                                   \
    _Pragma("unroll")                                                         \
    for (int i = 0; i < 8; ++i) {                                             \
        int idx = i * 32 + lane;                                              \
        int row = idx >> 4;                                                   \
        int col = (idx & 15) << 2;                                            \
        __builtin_amdgcn_global_load_async_to_lds_b128(                       \
            (as1_v4i_ptr)(Wnc + (size_t)(r0 + row) * K + (K0) + col),         \
            (as3_v4i_ptr)(&wt[wave][BUF][row * 64 + col]), 0, 0);             \
    }

    ISSUE_TILE(0, 0)
    for (int k0 = 0; k0 < K; k0 += 64) {
        const int buf = (k0 >> 6) & 1;
        if (k0 + 64 < K) {
            ISSUE_TILE(k0 + 64, buf ^ 1)
            __builtin_amdgcn_s_wait_asynccnt(8);   // current tile (in-order)
        } else {
            __builtin_amdgcn_s_wait_asynccnt(0);   // last tile
        }
        const float* tp = &wt[wave][buf][0];
#pragma unroll
        for (int kk = 0; kk < 64; kk += 4) {
            v2f a = *reinterpret_cast<const v2f*>(&tp[m * 64 + kk + off]);
            v2f b = *reinterpret_cast<const v2f*>(&xs[k0 + kk + off]);
            acc = __builtin_amdgcn_wmma_f32_16x16x4_f32(
                false, a, false, b, (short)0, acc, false, false);
        }
        // no barrier: tiles are wave-private; WMMA's register use forces the
        // dscnt wait before the buffer is re-issued (no WAR hazard)
    }
#undef ISSUE_TILE
#else
    for (int k0 = 0; k0 < K; k0 += 64) {
        // cooperative, coalesced synchronous staging of the 16x64 f32 tile
#pragma unroll
        for (int i = 0; i < 8; ++i) {
            int idx = i * 32 + lane;                // 0..255
            int row = idx >> 4;                     // 0..15
            int col = (idx & 15) << 2;              // 0,4,...,60
            const float4 v = *reinterpret_cast<const float4*>(
                &W[(size_t)(r0 + row) * K + k0 + col]);
            *reinterpret_cast<float4*>(&wt[wave][row * 64 + col]) = v;
            if (k0 + 64 < K) {  // prefetch next tile -> global_prefetch_b8
                __builtin_prefetch(&W[(size_t)(r0 + row) * K + k0 + 64 + col], 0, 0);
            }
        }
        __syncthreads();
#pragma unroll
        for (int kk = 0; kk < 64; kk += 4) {
            v2f a = *reinterpret_cast<const v2f*>(&wt[wave][m * 64 + kk + off]);
            v2f b = *reinterpret_cast<const v2f*>(&xs[k0 + kk + off]);
            acc = __builtin_amdgcn_wmma_f32_16x16x4_f32(
                false, a, false, b, (short)0, acc, false, false);
        }
        __syncthreads();
    }
#endif

    // D layout: VGPR v holds rows (r0+v) [lanes 0-15] and (r0+8+v) [lanes
    // 16-31]; all 16 columns identical -> lanes 0 and 16 write the 16 results.
    if (lane == 0 || lane == 16) {
        int rbase = r0 + ((lane >> 4) << 3);
#pragma unroll
        for (int v = 0; v < 8; ++v) {
            float val = acc[v] + bias[rbase + v];
            if (ACT_RELU) val = fmaxf(val, 0.0f);
            out[rbase + v] = val;
        }
    }
}

// ---------------------------------------------------------------------------
// K3: softmax over 512 logits; write weights to ws and to d_out tail.
// ---------------------------------------------------------------------------
__global__ void softmax512_kernel(const float* __restrict__ logits,
                                  float* __restrict__ aw,
                                  float* __restrict__ out_aw) {
    __shared__ float red[MAXLEN];
    int t = threadIdx.x;                 // 0..511
    float v = logits[t];

    red[t] = v;
    __syncthreads();
    for (int s = MAXLEN / 2; s > 0; s >>= 1) {
        if (t < s) red[t] = fmaxf(red[t], red[t + s]);
        __syncthreads();
    }
    float mx = red[0];
    __syncthreads();

    float e = expf(v - mx);
    red[t] = e;
    __syncthreads();
    for (int s = MAXLEN / 2; s > 0; s >>= 1) {
        if (t < s) red[t] += red[t + s];
        __syncthreads();
    }
    float w = e / red[0];
    aw[t]     = w;
    out_aw[t] = w;
}

// ---------------------------------------------------------------------------
// K4: attnApplied[n] = sum_k aw[k] * E[k][n]   (column reduction, coalesced)
// ---------------------------------------------------------------------------
__global__ void attn_apply_kernel(const float* __restrict__ E,
                                  const float* __restrict__ aw,
                                  float* __restrict__ dst) {
    __shared__ float a_s[MAXLEN];
    int t = threadIdx.x;
    for (int i = t; i < MAXLEN; i += 256) a_s[i] = aw[i];
    __syncthreads();

    int n = blockIdx.x * 256 + t;        // 0..1023
    float s = 0.0f;
#pragma unroll 8
    for (int k = 0; k < MAXLEN; ++k)
        s = fmaf(a_s[k], E[(size_t)k * H + n], s);
    dst[n] = s;
}

// ---------------------------------------------------------------------------
// K7: GRU elementwise (h0 == 0): gh = b_hh; h_new = (1-z)*n; out = sigmoid(h)
// d_out[0:H] = out, d_out[H:2H] = h_new.
// ---------------------------------------------------------------------------
__device__ __forceinline__ float sigmoidf(float x) {
    return 1.0f / (1.0f + expf(-x));
}

__global__ void gru_final_kernel(const float* __restrict__ gi,
                                 const float* __restrict__ b_hh,
                                 float* __restrict__ out) {
    int t = blockIdx.x * 256 + threadIdx.x;          // 0..1023
    float r = sigmoidf(gi[t]         + b_hh[t]);
    float z = sigmoidf(gi[H + t]     + b_hh[H + t]);
    float n = tanhf(gi[2 * H + t] + r * b_hh[2 * H + t]);
    float h = (1.0f - z) * n;
    out[t]     = sigmoidf(h);
    out[H + t] = h;
}

// ---------------------------------------------------------------------------
// Host-side launch chain (graph-capture safe: only kernel launches on stream)
// ---------------------------------------------------------------------------
extern "C" void kernel_launch(void* const* d_in, const int* in_sizes, int n_in,
                              void* d_out, int out_size, void* d_ws, size_t ws_size,
                              hipStream_t stream) {
    const int*   inputId     = (const int*)  d_in[0];
    const float* hidden      = (const float*)d_in[1];
    const float* encoderOuts = (const float*)d_in[2];
    const float* embed_table = (const float*)d_in[3];
    const float* attn_w      = (const float*)d_in[4];
    const float* attn_b      = (const float*)d_in[5];
    const float* comb_w      = (const float*)d_in[6];
    const float* comb_b      = (const float*)d_in[7];
    const float* w_ih        = (const float*)d_in[8];
    // d_in[9]  = w_hh : unused (h0 == 0)
    const float* b_ih        = (const float*)d_in[10];
    const float* b_hh        = (const float*)d_in[11];

    float* out = (float*)d_out;          // [out(1024) | h_new(1024) | attnW(512)]
    float* ws  = (float*)d_ws;

    float* c      = ws;                  // 2048
    float* logits = ws + 2048;           // 512
    float* aw     = ws + 2560;           // 512
    float* x2     = ws + 3072;           // 2048
    float* y      = ws + 5120;           // 1024
    float* gi     = ws + 6144;           // 3072

    // 1. c = [embed, hidden]; x2[0:H] = embed
    build_vecs_kernel<<<8, 256, 0, stream>>>(inputId, hidden, embed_table, c, x2);

    // 2. logits = c @ attn_w.T + attn_b        (N=512, K=2048) -> grid 4
    gemv_wmma_kernel<0><<<4, 256, 0, stream>>>(attn_w, c, attn_b, logits, 2 * H);

    // 3. softmax -> aw (ws) and d_out[2048:2560]
    softmax512_kernel<<<1, MAXLEN, 0, stream>>>(logits, aw, out + 2 * H);

    // 4. x2[H:2H] = aw @ encoderOuts           (coalesced column reduction)
    attn_apply_kernel<<<4, 256, 0, stream>>>(encoderOuts, aw, x2 + H);

    // 5. y = relu(x2 @ comb_w.T + comb_b)      (N=1024, K=2048) -> grid 8
    gemv_wmma_kernel<1><<<8, 256, 0, stream>>>(comb_w, x2, comb_b, y, 2 * H);

    // 6. gi = y @ w_ih.T + b_ih                (N=3072, K=1024) -> grid 24
    gemv_wmma_kernel<0><<<24, 256, 0, stream>>>(w_ih, y, b_ih, gi, H);

    // 7. GRU elementwise -> d_out[0:2048]
    gru_final_kernel<<<4, 256, 0, stream>>>(gi, b_hh, out);
}